// RelPositionMultiheadAttentionWeights_61400852464147
// MI455X (gfx1250) — compile-verified
//
#include <hip/hip_runtime.h>
#include <hip/hip_bf16.h>

typedef __attribute__((ext_vector_type(2))) float v2f;
typedef __attribute__((ext_vector_type(4))) float v4f;
typedef __attribute__((ext_vector_type(8))) float v8f;

// Problem constants
constexpr int S  = 1024;
constexpr int B  = 8;
constexpr int E  = 512;
constexpr int H  = 8;
constexpr int QD = 32;
constexpr int PD = 4;
constexpr int OPROJ = 544;           // (32*2+4)*8
constexpr int L  = 2 * S - 1;        // 2047

// ---------------------------------------------------------------------------
// Kernel A: proj = x @ W^T + b, scattered into q/k/p workspaces.
//   x: (S*B, 512) row-major (S,B,E contiguous -> row m = s*B+b)
//   w: (544, 512) row-major (so B-matrix B[K][N] = w[N][K] loads with the
//      same float2 pattern as A)
//   q/k ws: (H,B,S,32) f32 ; p ws: (H,B,S,4) f32
// One wave computes one 16x16 output tile with 128x v_wmma_f32_16x16x4_f32.
// ---------------------------------------------------------------------------
__global__ void __launch_bounds__(256)
in_proj_kernel(const float* __restrict__ x,
               const float* __restrict__ w,
               const float* __restrict__ bias,
               float* __restrict__ qws,
               float* __restrict__ kws,
               float* __restrict__ pws)
{
    const int lane = threadIdx.x & 31;
    const int wave = threadIdx.x >> 5;
    const int tile = blockIdx.x * 8 + wave;      // 512*34 = 17408 tiles total
    const int NT   = OPROJ / 16;                 // 34
    const int mt   = tile / NT;
    const int nt   = tile - mt * NT;
    const int m0   = mt * 16;
    const int n0   = nt * 16;

    const int r16  = lane & 15;
    const int half = lane >> 4;                  // 0 or 1
    const int kofs = half * 2;                   // f32 WMMA A/B: VGPR pair holds K={kofs,kofs+1}

    const float* arow = x + (m0 + r16) * E;      // A: q of GEMM (x rows)
    const float* brow = w + (n0 + r16) * E;      // B: w rows (transposed GEMM operand)

    v8f acc = {};
#pragma unroll 8
    for (int k0 = 0; k0 < E; k0 += 4) {
        v2f a  = *(const v2f*)(arow + k0 + kofs);
        v2f bm = *(const v2f*)(brow + k0 + kofs);
        acc = __builtin_amdgcn_wmma_f32_16x16x4_f32(
            /*neg_a=*/false, a, /*neg_b=*/false, bm,
            /*c_mod=*/(short)0, acc, /*reuse_a=*/false, /*reuse_b=*/false);
    }

    // C/D layout: VGPR r, lanes 0-15 -> M=r ; lanes 16-31 -> M=r+8 ; N = lane&15
    const int o    = n0 + r16;
    const float bv = bias[o];
#pragma unroll
    for (int r = 0; r < 8; ++r) {
        const int m = m0 + r + 8 * half;         // row of proj
        const int s = m >> 3;                    // /B (B==8)
        const int b = m & 7;
        const float v = acc[r] + bv;
        if (o < H * QD) {                        // q
            const int h = o >> 5, d = o & 31;
            qws[(((h * B + b) * S) + s) * QD + d] = v;
        } else if (o < 2 * H * QD) {             // k
            const int oo = o - H * QD;
            const int h = oo >> 5, d = oo & 31;
            kws[(((h * B + b) * S) + s) * QD + d] = v;
        } else {                                 // p
            const int oo = o - 2 * H * QD;
            const int h = oo >> 2, e = oo & 3;
            pws[(((h * B + b) * S) + s) * PD + e] = v;
        }
    }
}

// ---------------------------------------------------------------------------
// Kernel B: pe[t][o] = sum_d pos_emb[t][d] * lpw[o][d]   (2047 x 32, K=192)
// ---------------------------------------------------------------------------
__global__ void __launch_bounds__(256)
pe_kernel(const float* __restrict__ pos_emb,    // (2047,192)
          const float* __restrict__ lpw,        // (32,192)
          float* __restrict__ pe)               // (2047,32)
{
    const int idx = blockIdx.x * blockDim.x + threadIdx.x;
    if (idx >= L * (H * PD)) return;
    const int t = idx >> 5;
    const int o = idx & 31;
    const float* a  = pos_emb + t * 192;
    const float* wv = lpw + o * 192;
    float s = 0.f;
#pragma unroll 4
    for (int d = 0; d < 192; ++d) s += a[d] * wv[d];
    pe[idx] = s;
}

// ---------------------------------------------------------------------------
// Kernel C: per (h,b,16-row tile): scores = qk^T + rel-shift(pos), mask,
// softmax over 1024 cols, store. 8 waves; wave w owns columns [128w,128w+128).
// ---------------------------------------------------------------------------
constexpr int ROWSTRIDE = S + 8;                 // LDS row padding (bank-conflict break)

__global__ void __launch_bounds__(256)
attn_kernel(const float* __restrict__ qws,      // (H,B,S,32)
            const float* __restrict__ kws,      // (H,B,S,32)
            const float* __restrict__ pws,      // (H,B,S,4)
            const float* __restrict__ pe,       // (2047,32)
            const unsigned char* __restrict__ kmask, // (B,S) bool
            float* __restrict__ out)            // (H,B,S,S)
{
    __shared__ float sc[16 * ROWSTRIDE];         // ~66 KB

    const int bid = blockIdx.x;                  // H*B*(S/16) = 4096
    const int it  = bid & 63;
    const int b   = (bid >> 6) & 7;
    const int h   = bid >> 9;
    const int i0  = it * 16;

    const int lane = threadIdx.x & 31;
    const int wave = threadIdx.x >> 5;
    const int r16  = lane & 15;
    const int half = lane >> 4;
    const int kofs = half * 2;

    const float* qh = qws + (size_t)(h * B + b) * S * QD;
    const float* kh = kws + (size_t)(h * B + b) * S * QD;
    const float* ph = pws + (size_t)(h * B + b) * S * PD;
    const float* peh = pe + h * PD;              // column block of pe row

    // Preload A (q tile, 16x32) in WMMA A layout: aq[kk] covers K = 4kk..4kk+3
    v2f aq[8];
    {
        const float* qrow = qh + (i0 + r16) * QD;
#pragma unroll
        for (int kk = 0; kk < 8; ++kk)
            aq[kk] = *(const v2f*)(qrow + kk * 4 + kofs);
    }
    // Preload p rows in accumulator-row order: pv[r] is row i0 + r + 8*half
    v4f pv[8];
#pragma unroll
    for (int r = 0; r < 8; ++r)
        pv[r] = *(const v4f*)(ph + (i0 + r + 8 * half) * PD);

#pragma unroll 1
    for (int t = 0; t < 8; ++t) {
        const int j0 = (wave * 8 + t) * 16;
        const int j  = j0 + r16;

        // Positional scores straight into the WMMA accumulator layout:
        // element (M,N): rel = (S-1) + (j0+N) - (i0+M), always in [0, 2S-2]
        v8f acc;
#pragma unroll
        for (int r = 0; r < 8; ++r) {
            const int i   = i0 + r + 8 * half;
            const int rel = (S - 1) + j - i;
            const v4f pr  = *(const v4f*)(peh + rel * (H * PD));
            acc[r] = pv[r].x * pr.x + pv[r].y * pr.y + pv[r].z * pr.z + pv[r].w * pr.w;
        }

        // q @ k^T for this 16x16 tile: 8 fp32 WMMAs over K=32
        const float* krow = kh + (j0 + r16) * QD;
#pragma unroll
        for (int kk = 0; kk < 8; ++kk) {
            v2f bm = *(const v2f*)(krow + kk * 4 + kofs);
            acc = __builtin_amdgcn_wmma_f32_16x16x4_f32(
                false, aq[kk], false, bm, (short)0, acc, false, false);
        }

        // key padding mask (column-wise)
        const bool masked = kmask[b * S + j] != 0;
#pragma unroll
        for (int r = 0; r < 8; ++r) {
            float v = masked ? -__builtin_inff() : acc[r];
            sc[(r + 8 * half) * ROWSTRIDE + j] = v;
        }
    }

    __syncthreads();

    // Softmax: wave w handles rows 2w, 2w+1; 32 lanes sweep 1024 cols.
#pragma unroll
    for (int rr = 0; rr < 2; ++rr) {
        const int row = wave * 2 + rr;
        float* srow = sc + row * ROWSTRIDE;

        float m = -__builtin_inff();
        for (int c = lane; c < S; c += 32) m = fmaxf(m, srow[c]);
#pragma unroll
        for (int off = 16; off > 0; off >>= 1)
            m = fmaxf(m, __shfl_xor(m, off, 32));

        float sum = 0.f;
        for (int c = lane; c < S; c += 32) {
            const float e = __expf(srow[c] - m);
            srow[c] = e;
            sum += e;
        }
#pragma unroll
        for (int off = 16; off > 0; off >>= 1)
            sum += __shfl_xor(sum, off, 32);
        const float inv = 1.0f / sum;

        float* orow = out + ((size_t)(h * B + b) * S + (i0 + row)) * S;
        for (int c = lane; c < S; c += 32) orow[c] = srow[c] * inv;
    }
}

// ---------------------------------------------------------------------------
// Launch
// ---------------------------------------------------------------------------
extern "C" void kernel_launch(void* const* d_in, const int* in_sizes, int n_in,
                              void* d_out, int out_size, void* d_ws, size_t ws_size,
                              hipStream_t stream) {
    const float* x        = (const float*)d_in[0];   // (S,B,512)
    const float* pos_emb  = (const float*)d_in[1];   // (1,2047,192)
    const unsigned char* kmask = (const unsigned char*)d_in[2]; // (B,S) bool
    const float* in_proj_w = (const float*)d_in[3];  // (544,512)
    const float* in_proj_b = (const float*)d_in[4];  // (544)
    const float* lpw       = (const float*)d_in[5];  // (32,192)
    float* out = (float*)d_out;

    float* ws  = (float*)d_ws;
    float* qws = ws;                                  // H*B*S*32 = 2,097,152 f
    float* kws = qws + (size_t)H * B * S * QD;        // + 2,097,152 f
    float* pws = kws + (size_t)H * B * S * QD;        // + 262,144 f
    float* pe  = pws + (size_t)H * B * S * PD;        // + 65,504 f  (~17.4 MB total)

    // A: 512*34 = 17408 wave-tiles / 8 waves per block
    in_proj_kernel<<<2176, 256, 0, stream>>>(x, in_proj_w, in_proj_b, qws, kws, pws);
    // B: 2047*32 outputs
    pe_kernel<<<(L * H * PD + 255) / 256, 256, 0, stream>>>(pos_emb, lpw, pe);
    // C: H*B*(S/16) = 4096 blocks
    attn_kernel<<<H * B * (S / 16), 256, 0, stream>>>(qws, kws, pws, pe, kmask, out);
}